// SparseAttention_41394894799657
// MI455X (gfx1250) — compile-verified
//
#include <hip/hip_runtime.h>

// Restormer-style channel attention on MI455X (gfx1250), fp32 end-to-end.
// All GEMMs use V_WMMA_F32_16X16X4_F32 (CDNA5 fp32 WMMA, K=4 steps).
// Fragments are loaded as aligned v2f pairs (-> single ds_load_b64 into an
// even-aligned VGPR pair, no v_mov shuffles before the WMMA).

typedef float v2f __attribute__((ext_vector_type(2)));
typedef float v8f __attribute__((ext_vector_type(8)));

#define BATCH 8
#define CDIM  192
#define C3    576
#define HWN   16384
#define HEADS 4
#define CH    48

__device__ __forceinline__ v8f wmma4(v2f a, v2f b, v8f c) {
  // 8 args: (neg_a, A, neg_b, B, c_mod, C, reuse_a, reuse_b)
  return __builtin_amdgcn_wmma_f32_16x16x4_f32(false, a, false, b, (short)0, c,
                                               false, false);
}

__device__ __forceinline__ v8f vzero8() {
  v8f z;
#pragma unroll
  for (int i = 0; i < 8; ++i) z[i] = 0.0f;
  return z;
}

__device__ __forceinline__ v2f ld2(const float* p) {
  return *(const v2f*)p;  // 8B-aligned by construction (even k, even pitch)
}

// D[b][m][n] = sum_k Wm[m][k] * X[b][k][n],  K == 192, N == 16384.
// Block: 256 thr = 8 waves; tile 64(M) x 128(N); K staged in LDS 64-chunks.
// X tile stored TRANSPOSED in LDS ([n][k]) so B-fragments are contiguous
// K-pairs -> one conflict-free ds_load_b64 each.
__global__ __launch_bounds__(256) void gemm_k192(
    const float* __restrict__ Wm, const float* __restrict__ X,
    float* __restrict__ D, int M) {
  __shared__ float xst[128][68];  // [n][k], pitch 68 (272B, 8B aligned)
  __shared__ float wsh[64][68];   // [m][k], pitch 68
  const int tid  = threadIdx.x;
  const int lane = tid & 31;
  const int wave = tid >> 5;
  const int lr   = lane & 15;
  const int hi   = lane >> 4;          // 0: K+0/1, 1: K+2/3
  const int wm   = wave >> 1;          // 0..3  (M group)
  const int wn   = wave & 1;           // 0..1  (N half)
  const int b    = blockIdx.z;
  const int nblk = blockIdx.x * 128;
  const int mblk = blockIdx.y * 64;
  const float* Xb = X + (size_t)b * CDIM * HWN;

  v8f acc[4];
#pragma unroll
  for (int t = 0; t < 4; ++t) acc[t] = vzero8();

  for (int k0 = 0; k0 < 192; k0 += 64) {
    // Coalesced global read (consecutive tid -> consecutive n), transposed
    // LDS write (2-way bank conflict, read path is conflict-free).
    for (int idx = tid; idx < 64 * 128; idx += 256) {
      int k = idx >> 7, n = idx & 127;
      xst[n][k] = Xb[(size_t)(k0 + k) * HWN + nblk + n];
    }
    for (int idx = tid; idx < 64 * 64; idx += 256) {
      int m = idx >> 6, k = idx & 63;
      wsh[m][k] = Wm[(size_t)(mblk + m) * 192 + k0 + k];
    }
    // Prefetch next K-chunk of X while this one is consumed.
    if (k0 + 64 < 192) {
      int k = tid >> 2, n = (tid & 3) * 32;  // 64 rows x 4 lines
      __builtin_prefetch(&Xb[(size_t)(k0 + 64 + k) * HWN + nblk + n], 0, 0);
    }
    __syncthreads();
#pragma unroll
    for (int kk = 0; kk < 64; kk += 4) {
      v2f a = ld2(&wsh[wm * 16 + lr][kk + 2 * hi]);
#pragma unroll
      for (int t = 0; t < 4; ++t) {
        v2f bb = ld2(&xst[wn * 64 + t * 16 + lr][kk + 2 * hi]);
        acc[t] = wmma4(a, bb, acc[t]);
      }
    }
    __syncthreads();
  }

  float* Db = D + (size_t)b * M * HWN;
#pragma unroll
  for (int t = 0; t < 4; ++t)
#pragma unroll
    for (int r = 0; r < 8; ++r) {
      int m = mblk + wm * 16 + r + 8 * hi;  // C/D layout: vgpr r -> M=r / r+8
      int n = nblk + wn * 64 + t * 16 + lr;
      Db[(size_t)m * HWN + n] = acc[t][r];
    }
}

// 3x3 depthwise conv, SAME padding, groups = 576.
__global__ __launch_bounds__(256) void dwconv3(const float* __restrict__ in,
                                               const float* __restrict__ wdw,
                                               float* __restrict__ out) {
  size_t idx = (size_t)blockIdx.x * 256 + threadIdx.x;
  int x = (int)(idx & 127);
  int y = (int)((idx >> 7) & 127);
  size_t bc = idx >> 14;  // b*576 + c
  int c = (int)(bc % C3);
  const float* w9 = wdw + c * 9;
  const float* p  = in + (bc << 14);
  float s = 0.0f;
#pragma unroll
  for (int dy = -1; dy <= 1; ++dy) {
    int yy = y + dy;
    if (yy < 0 || yy > 127) continue;
#pragma unroll
    for (int dx = -1; dx <= 1; ++dx) {
      int xx = x + dx;
      if (xx < 0 || xx > 127) continue;
      s += p[yy * 128 + xx] * w9[(dy + 1) * 3 + (dx + 1)];
    }
  }
  out[idx] = s;
}

// Row L2 norms of q (rows 0..191 of qkvd per batch) and k (rows 192..383).
// Writes scale = 1 / max(||row||, 1e-12).
__global__ __launch_bounds__(256) void rownorm(const float* __restrict__ qkvd,
                                               float* __restrict__ qs,
                                               float* __restrict__ ks) {
  __shared__ float red[256];
  int row = blockIdx.x;                      // 0..3071
  int isK = row >= BATCH * CDIM;
  int r   = isK ? row - BATCH * CDIM : row;  // b*192 + c
  int b = r / CDIM, c = r % CDIM;
  const float* p = qkvd + ((size_t)b * C3 + (isK ? CDIM : 0) + c) * HWN;
  float s = 0.0f;
  for (int i = threadIdx.x; i < HWN; i += 256) {
    float v = p[i];
    s += v * v;
  }
  red[threadIdx.x] = s;
  __syncthreads();
  for (int off = 128; off > 0; off >>= 1) {
    if (threadIdx.x < off) red[threadIdx.x] += red[threadIdx.x + off];
    __syncthreads();
  }
  if (threadIdx.x == 0) {
    float n = sqrtf(red[0]);
    (isK ? ks : qs)[r] = 1.0f / fmaxf(n, 1e-12f);
  }
}

// attn[b,h] = relu( diag(qs) (q k^T) diag(ks) * tau ),  48x48, K = 16384.
// 288 thr = 9 waves, one 16x16 tile per wave; K streamed via LDS 64-chunks.
__global__ __launch_bounds__(288) void attn_wmma(
    const float* __restrict__ qkvd, const float* __restrict__ qs,
    const float* __restrict__ ks, const float* __restrict__ temp,
    float* __restrict__ attn) {
  __shared__ float qlds[48][68];
  __shared__ float klds[48][68];
  const int tid  = threadIdx.x;
  const int lane = tid & 31;
  const int wave = tid >> 5;  // 0..8
  const int lr = lane & 15, hi = lane >> 4;
  const int ti = wave / 3, tj = wave % 3;
  const int bh = blockIdx.x;  // b*4 + h
  const int b = bh >> 2, h = bh & 3;
  const float* qb = qkvd + ((size_t)b * C3 + h * CH) * HWN;
  const float* kb = qkvd + ((size_t)b * C3 + CDIM + h * CH) * HWN;

  v8f acc = vzero8();
  for (int n0 = 0; n0 < HWN; n0 += 64) {
    for (int idx = tid; idx < 48 * 64; idx += 288) {
      int r = idx >> 6, cc = idx & 63;
      qlds[r][cc] = qb[(size_t)r * HWN + n0 + cc];
      klds[r][cc] = kb[(size_t)r * HWN + n0 + cc];
    }
    // Prefetch next chunk (one line per q/k row; 288 thr cover 2*48 rows+).
    if (n0 + 64 < HWN && tid < 96) {
      const float* base = (tid < 48) ? qb : kb;
      int r = (tid < 48) ? tid : tid - 48;
      __builtin_prefetch(&base[(size_t)r * HWN + n0 + 64], 0, 0);
    }
    __syncthreads();
#pragma unroll
    for (int kk = 0; kk < 64; kk += 4) {
      v2f a  = ld2(&qlds[ti * 16 + lr][kk + 2 * hi]);
      v2f bb = ld2(&klds[tj * 16 + lr][kk + 2 * hi]);
      acc = wmma4(a, bb, acc);
    }
    __syncthreads();
  }

  float tau = temp[h];
#pragma unroll
  for (int r = 0; r < 8; ++r) {
    int c = ti * 16 + r + 8 * hi;
    int d = tj * 16 + lr;
    float v = acc[r] * qs[b * CDIM + h * CH + c] * ks[b * CDIM + h * CH + d] * tau;
    attn[((size_t)bh * CH + c) * CH + d] = fmaxf(v, 0.0f);
  }
}

// y[b,h] = attn(48x48) @ v(48x16384).  Block: 256 thr, 128 pixels; each wave
// owns 16 pixels x all 3 M-tiles; attn held in LDS, v streamed from global.
__global__ __launch_bounds__(256) void av_wmma(const float* __restrict__ attn,
                                               const float* __restrict__ qkvd,
                                               float* __restrict__ y) {
  __shared__ float alds[48][52];  // pitch 52 (208B): b64 reads conflict-free
  const int tid  = threadIdx.x;
  const int lane = tid & 31, wave = tid >> 5;
  const int lr = lane & 15, hi = lane >> 4;
  const int bh = blockIdx.y, b = bh >> 2, h = bh & 3;
  const int n0 = blockIdx.x * 128 + wave * 16;

  for (int idx = tid; idx < 48 * 48; idx += 256) {
    int r = idx / 48, cc = idx % 48;
    alds[r][cc] = attn[(size_t)bh * CH * CH + idx];
  }
  __syncthreads();

  const float* vb = qkvd + ((size_t)b * C3 + 2 * CDIM + h * CH) * HWN;
  v8f acc[3];
#pragma unroll
  for (int t = 0; t < 3; ++t) acc[t] = vzero8();

#pragma unroll
  for (int kk = 0; kk < 48; kk += 4) {
    v2f bb;  // spans two v rows (stride HWN): two b32 global loads
    bb.x = vb[(size_t)(kk + 2 * hi) * HWN + n0 + lr];
    bb.y = vb[(size_t)(kk + 2 * hi + 1) * HWN + n0 + lr];
#pragma unroll
    for (int mt = 0; mt < 3; ++mt) {
      v2f a = ld2(&alds[mt * 16 + lr][kk + 2 * hi]);
      acc[mt] = wmma4(a, bb, acc[mt]);
    }
  }

  float* yb = y + ((size_t)b * CDIM + h * CH) * HWN;
#pragma unroll
  for (int mt = 0; mt < 3; ++mt)
#pragma unroll
    for (int r = 0; r < 8; ++r) {
      int c = mt * 16 + r + 8 * hi;
      yb[(size_t)c * HWN + n0 + lr] = acc[mt][r];
    }
}

extern "C" void kernel_launch(void* const* d_in, const int* in_sizes, int n_in,
                              void* d_out, int out_size, void* d_ws,
                              size_t ws_size, hipStream_t stream) {
  const float* x      = (const float*)d_in[0];  // (8,192,128,128)
  const float* w_qkv  = (const float*)d_in[1];  // (576,192)
  const float* w_dw   = (const float*)d_in[2];  // (576,1,3,3)
  const float* w_proj = (const float*)d_in[3];  // (192,192)
  const float* temp   = (const float*)d_in[4];  // (4,1,1)

  const size_t big = (size_t)BATCH * C3 * HWN;  // 75,497,472 floats
  float* qkv  = (float*)d_ws;                   // 1x1 conv output; reused as y
  float* qkvd = qkv + big;                      // after depthwise conv
  float* attn = qkvd + big;                     // 32 * 48 * 48
  float* qs   = attn + (size_t)BATCH * HEADS * CH * CH;
  float* ks   = qs + BATCH * CDIM;
  float* y    = qkv;                            // safe reuse after dwconv3

  // 1) qkv = W_qkv @ x  (per batch GEMM, M=576, K=192, N=16384)
  gemm_k192<<<dim3(HWN / 128, C3 / 64, BATCH), 256, 0, stream>>>(
      w_qkv, x, qkv, C3);

  // 2) 3x3 depthwise conv
  dwconv3<<<(unsigned)(big / 256), 256, 0, stream>>>(qkv, w_dw, qkvd);

  // 3) row norms of q and k
  rownorm<<<2 * BATCH * CDIM, 256, 0, stream>>>(qkvd, qs, ks);

  // 4) attn = relu(diag(qs) q k^T diag(ks) * tau)
  attn_wmma<<<BATCH * HEADS, 288, 0, stream>>>(qkvd, qs, ks, temp, attn);

  // 5) y = attn @ v
  av_wmma<<<dim3(HWN / 128, BATCH * HEADS), 256, 0, stream>>>(attn, qkvd, y);

  // 6) out = W_proj @ y  (M=192, K=192, N=16384)
  gemm_k192<<<dim3(HWN / 128, CDIM / 64, BATCH), 256, 0, stream>>>(
      w_proj, y, (float*)d_out, CDIM);
}